// ConvQuadInterp3d_82738249990632
// MI455X (gfx1250) — compile-verified
//
#include <hip/hip_runtime.h>

// ConvQuadInterp3d: 27-pt stencil, strict local-max detect, 3x3 Newton refine.
// B=2, CH=1, D=10, H=512, W=512.
// Outputs: coords (B,1,D,3,H,W) ++ y_max (B,1,D,H,W), fp32.
//
// Memory bound (~21MB in, ~84MB out -> ~4.5us floor @ 23.3 TB/s).
// Strategy: one block per (b, h-tile, w-tile), loop over d with a 4-slot
// rotating LDS slice ring filled by CDNA5 async global->LDS DMA
// (global_load_async_to_lds_b32 / ASYNCcnt), prefetching slice d+2 while
// computing slice d. Input slices are loaded once (+2 clamped) instead of 3x.

#define B_   2
#define D_   10
#define H_   512
#define W_   512
#define BONUS 10.0f

#define TW 32
#define TH 8
#define HALO_W (TW + 2)              // 34
#define HALO_H (TH + 2)              // 10
#define SLICE  (HALO_H * HALO_W)     // 340 floats per slice
#define NBUF   4                     // rotating ring (compute uses 3, prefetch 1)

// Wave-relative LDS byte offset: generic (flat) addresses of __shared__ data
// carry the LDS offset in the low 32 bits (ISA 10.2 aperture mapping).
__device__ __forceinline__ unsigned lds_off(const void* p) {
    return (unsigned)(unsigned long long)p;
}

__device__ __forceinline__ void async_g2l_b32(const float* gsrc, unsigned ldsByteOff) {
    // CDNA5 async DMA: copies 4 bytes from the global address in vaddr to the LDS address in vdst; tracked on ASYNCcnt.
    asm volatile("global_load_async_to_lds_b32 %0, %1, off"
                 :: "v"(ldsByteOff), "v"(gsrc)
                 : "memory");
}

__device__ __forceinline__ void wait_async0() {
    asm volatile("s_wait_asynccnt 0" ::: "memory");
}

__global__ __launch_bounds__(256)
void quad_interp3d_kernel(const float* __restrict__ x,
                          float* __restrict__ coords,   // (B*D, 3, H, W)
                          float* __restrict__ ymax)     // (B*D, H, W)
{
    __shared__ float tile[NBUF][SLICE];

    const int tid = threadIdx.x;
    const int b   = blockIdx.z;              // batch
    const int h0  = blockIdx.y * TH;
    const int w0  = blockIdx.x * TW;

    const size_t hw = (size_t)H_ * W_;
    const float* xb = x + (size_t)b * D_ * hw;

    // Per-thread halo-load source offsets (h/w clamp is d-invariant).
    // Element A: idx = tid; element B: idx = tid + 256 (valid iff tid < 84).
    const int sxA = tid % HALO_W;
    const int syA = tid / HALO_W;
    const size_t offA = (size_t)min(max(h0 - 1 + syA, 0), H_ - 1) * W_
                      +         min(max(w0 - 1 + sxA, 0), W_ - 1);
    const int  idxB = tid + 256;
    const bool hasB = idxB < SLICE;          // tid < 84
    const int sxB = idxB % HALO_W;
    const int syB = idxB / HALO_W;
    const size_t offB = (size_t)min(max(h0 - 1 + syB, 0), H_ - 1) * W_
                      +         min(max(w0 - 1 + sxB, 0), W_ - 1);

    // Buffer (z+1)&3 holds (depth-clamped) slice z.
    auto load_slice = [&](int z, int buf) {
        const float* sbase = xb + (size_t)min(max(z, 0), D_ - 1) * hw;
        async_g2l_b32(sbase + offA, lds_off(&tile[buf][tid]));
        if (hasB)
            async_g2l_b32(sbase + offB, lds_off(&tile[buf][idxB]));
    };

    // Preload slices z = -1, 0, 1 into buffers 0, 1, 2.
    load_slice(-1, 0);
    load_slice( 0, 1);
    load_slice( 1, 2);

    const int ty = tid >> 5;     // 0..7
    const int tx = tid & 31;     // lane -> contiguous w (coalesced stores)
    const int h  = h0 + ty;
    const int w  = w0 + tx;
    const size_t pix = (size_t)h * W_ + w;

    for (int d = 0; d < D_; ++d) {
        // Drain the prefetch issued last iteration (it overlapped compute),
        // then barrier: all waves' LDS writes visible + all waves done
        // reading the buffer we are about to overwrite.
        wait_async0();
        __syncthreads();

        // Prefetch slice d+2 into the slot that held slice d-2.
        if (d < D_ - 1)
            load_slice(d + 2, (d + 3) & 3);

        const float* s0 = tile[(d)     & 3];   // slice d-1
        const float* s1 = tile[(d + 1) & 3];   // slice d
        const float* s2 = tile[(d + 2) & 3];   // slice d+1

        float v[3][3][3];
#pragma unroll
        for (int yy = 0; yy < 3; ++yy) {
            const int r = (ty + yy) * HALO_W + tx;
#pragma unroll
            for (int xx = 0; xx < 3; ++xx) {
                v[0][yy][xx] = s0[r + xx];
                v[1][yy][xx] = s1[r + xx];
                v[2][yy][xx] = s2[r + xx];
            }
        }

        const float cc = v[1][1][1];

        float nmax = 0.0f;       // max over 26 neighbors, floored at 0
#pragma unroll
        for (int z = 0; z < 3; ++z)
#pragma unroll
            for (int yy = 0; yy < 3; ++yy)
#pragma unroll
                for (int xx = 0; xx < 3; ++xx)
                    if (!(z == 1 && yy == 1 && xx == 1))
                        nmax = fmaxf(nmax, v[z][yy][xx]);

        const bool m = cc > nmax;

        // gradients / Hessian (central differences on replicate-padded field)
        const float gx  = 0.5f  * (v[1][1][2] - v[1][1][0]);
        const float gy  = 0.5f  * (v[1][2][1] - v[1][0][1]);
        const float gs  = 0.5f  * (v[2][1][1] - v[0][1][1]);
        const float dxx = v[1][1][0] + v[1][1][2] - 2.0f * cc;
        const float dyy = v[1][0][1] + v[1][2][1] - 2.0f * cc;
        const float dss = v[0][1][1] + v[2][1][1] - 2.0f * cc;
        const float dxy = 0.25f * (v[1][0][0] + v[1][2][2] - v[1][2][0] - v[1][0][2]);
        const float dys = 0.25f * (v[0][0][1] + v[2][2][1] - v[2][0][1] - v[0][2][1]);
        const float dxs = 0.25f * (v[0][1][0] + v[2][1][2] - v[2][1][0] - v[0][1][2]);

        // where(m, H, I) / where(m, g, 0) -> branchless 3x3 cofactor solve
        const float h00 = m ? dxx : 1.0f;
        const float h01 = m ? dxy : 0.0f;
        const float h02 = m ? dxs : 0.0f;
        const float h10 = h01;
        const float h11 = m ? dyy : 1.0f;
        const float h12 = m ? dys : 0.0f;
        const float h20 = h02;
        const float h21 = h12;
        const float h22 = m ? dss : 1.0f;
        const float b0  = m ? gx : 0.0f;
        const float b1  = m ? gy : 0.0f;
        const float b2  = m ? gs : 0.0f;

        const float c00 = h11 * h22 - h12 * h21;
        const float c01 = h10 * h22 - h12 * h20;
        const float c02 = h10 * h21 - h11 * h20;
        const float det = h00 * c00 - h01 * c01 + h02 * c02;
        const float inv = 1.0f / det;

        const float t0 = b1 * h22 - h12 * b2;
        const float t1 = b1 * h21 - h11 * b2;
        const float t2 = h10 * b2 - b1 * h20;

        const float sx  = (b0 * c00 - h01 * t0 + h02 * t1) * inv;
        const float sy  = (h00 * t0 - b0 * c01 + h02 * t2) * inv;
        const float ssv = (-h00 * t1 - h01 * t2 + b0 * c02) * inv;

        float dx0 = -sx, dx1 = -sy, dx2 = -ssv;
        const bool far =
            fmaxf(fmaxf(fabsf(dx0), fabsf(dx1)), fabsf(dx2)) > 0.7f;
        if (far) { dx0 = 0.0f; dx1 = 0.0f; dx2 = 0.0f; }

        const float dyv = 0.5f * (gx * dx0 + gy * dx1 + gs * dx2);

        // ---- coalesced stores: 3 coord planes + y_max ----
        const int plane = b * D_ + d;
        const size_t cbase = (size_t)plane * 3 * hw + pix;
        coords[cbase]          = (float)d + dx2;
        coords[cbase + hw]     = (float)h + dx1;
        coords[cbase + 2 * hw] = (float)w + dx0;
        ymax[(size_t)plane * hw + pix] = cc + dyv + (m ? BONUS : 0.0f);
    }
}

extern "C" void kernel_launch(void* const* d_in, const int* in_sizes, int n_in,
                              void* d_out, int out_size, void* d_ws, size_t ws_size,
                              hipStream_t stream) {
    const float* x = (const float*)d_in[0];
    float* out    = (float*)d_out;
    float* coords = out;                                    // B*D*3*H*W floats
    float* ymax   = out + (size_t)B_ * D_ * 3 * H_ * W_;    // B*D*H*W floats

    dim3 grid(W_ / TW, H_ / TH, B_);   // (16, 64, 2), each block loops d=0..9
    quad_interp3d_kernel<<<grid, 256, 0, stream>>>(x, coords, ymax);
}